// SSIMLoss_69406671503998
// MI455X (gfx1250) — compile-verified
//
#include <hip/hip_runtime.h>
#include <stdint.h>

// SSIM loss, fused single pass.
// x, y: f32[32,1,1024,1024]. Output: scalar f32 = 1 - mean(ssim_map).
// Memory-bound: 268 MB in / 4 B out -> ~11.5 us floor at 23.3 TB/s. Each input
// element is pulled once through the CDNA5 async global->LDS copy engine
// (ASYNCcnt-tracked, no VGPR staging), the five 3x3 box sums are formed from
// LDS with a 3-row sliding window, and the mean is reduced with one f64 atomic
// per block.

#define IMG_W   1024
#define IMG_H   1024
#define N_IMG   32
#define IMG_HW  (IMG_W * IMG_H)

#define TILE_W  128
#define TILE_H  32
#define LDS_STRIDE 136          // floats per LDS row (34 x 16B groups, 16B aligned)
#define LDS_ROWS   34           // TILE_H + 2 halo rows
#define LDS_GRPS   34           // 16B groups per row
#define LDS_TILE   (LDS_ROWS * LDS_STRIDE)   // 4624 floats per array

// CDNA5 async copy: global memory -> LDS, 16 bytes per lane, tracked by
// ASYNCcnt. GVS addressing: mem = SGPR64 base + VGPR32 byte offset. VDST holds
// the LDS byte address (low 32 bits of generic shared pointer = LDS offset).
__device__ __forceinline__ void async_copy16_to_lds(uint32_t lds_byte_addr,
                                                    uint32_t glb_byte_off,
                                                    const float* base) {
    asm volatile("global_load_async_to_lds_b128 %0, %1, %2"
                 :
                 : "v"(lds_byte_addr), "v"(glb_byte_off),
                   "s"((unsigned long long)(uintptr_t)base)
                 : "memory");
}

__device__ __forceinline__ void wait_async_all() {
    asm volatile("s_wait_asynccnt 0x0" ::: "memory");
}

__global__ void __launch_bounds__(256)
ssim_main_kernel(const float* __restrict__ x, const float* __restrict__ y,
                 double* __restrict__ g_acc) {
    __shared__ __align__(16) float smem[2 * LDS_TILE];
    __shared__ float red[8];

    float* xs = smem;
    float* ys = smem + LDS_TILE;

    const int tid = threadIdx.x;
    const int c0  = blockIdx.x * TILE_W;   // tile's first output column
    const int r0  = blockIdx.y * TILE_H;   // tile's first output row
    const int n   = blockIdx.z;            // image index
    const int cbase = c0 - 4;              // global col of LDS col 0 (16B aligned)
    const int rbase = r0 - 1;              // global row of LDS row 0

    const uint32_t xs_lds = (uint32_t)(uintptr_t)xs;
    const uint32_t ys_lds = (uint32_t)(uintptr_t)ys;
    const uint32_t img_off = (uint32_t)n * (uint32_t)IMG_HW;

    // Block-uniform: does this tile's halo cross the image border?
    const bool edge = (blockIdx.x == 0) | (blockIdx.x == gridDim.x - 1) |
                      (blockIdx.y == 0) | (blockIdx.y == gridDim.y - 1);

    if (!edge) {
        // Interior fast path: every 16B group is in-bounds -> no zero-fill, no
        // extra barrier, unconditional full-EXEC async issues.
        // li = lr * 34 + g, walked without division: thread tid starts at
        // (tid/34, tid%34) and advances g by 256 mod 34 each step.
        int lr = tid / LDS_GRPS;
        int g  = tid - lr * LDS_GRPS;
#pragma unroll
        for (int it = 0; it < 5; ++it) {            // ceil(34*34/256) = 5; last partial
            if (lr < LDS_ROWS) {
                const uint32_t goff =
                    (img_off + (uint32_t)(rbase + lr) * IMG_W +
                     (uint32_t)(cbase + g * 4)) * 4u;
                const uint32_t loff = (uint32_t)(lr * LDS_STRIDE + g * 4) * 4u;
                async_copy16_to_lds(xs_lds + loff, goff, x);
                async_copy16_to_lds(ys_lds + loff, goff, y);
            }
            // advance by 256 load-slots: 256 = 7*34 + 18
            lr += 7;
            g  += 18;
            if (g >= LDS_GRPS) { g -= LDS_GRPS; lr += 1; }
        }
    } else {
        // Border path: zero-fill implements the conv's zero padding, then load
        // only the in-image groups (EXEC-predicated per lane).
        for (int i = tid; i < 2 * LDS_TILE; i += 256) smem[i] = 0.0f;
        __syncthreads();
        for (int li = tid; li < LDS_ROWS * LDS_GRPS; li += 256) {
            const int lr = li / LDS_GRPS;
            const int g  = li - lr * LDS_GRPS;
            const int grow = rbase + lr;
            const int gcol = cbase + g * 4;  // 16B-aligned; width%4==0 -> group all-in or all-out
            if ((unsigned)grow < IMG_H && (unsigned)gcol < IMG_W) {
                const uint32_t goff =
                    (img_off + (uint32_t)grow * IMG_W + (uint32_t)gcol) * 4u;
                const uint32_t loff = (uint32_t)(lr * LDS_STRIDE + g * 4) * 4u;
                async_copy16_to_lds(xs_lds + loff, goff, x);
                async_copy16_to_lds(ys_lds + loff, goff, y);
            }
        }
    }
    wait_async_all();
    __syncthreads();

    // Each thread owns one column, 16 consecutive output rows.
    const int tc  = tid & (TILE_W - 1);        // 0..127
    const int tr0 = (tid >> 7) * 16;           // 0 or 16

    const float inv9 = 1.0f / 9.0f;
    const float C1f  = 1e-4f;                  // 0.01^2
    const float C2f  = 9e-4f;                  // 0.03^2
    const float EPSf = 1e-8f;

    // Horizontal 3-wide sums for the 5 channels at LDS row lr, column tc.
    auto rowsum = [&](int lr, float& hx, float& hy,
                      float& hxx, float& hyy, float& hxy) {
        const float* xr = xs + lr * LDS_STRIDE + tc + 3;
        const float* yr = ys + lr * LDS_STRIDE + tc + 3;
        const float x0 = xr[0], x1 = xr[1], x2 = xr[2];
        const float y0 = yr[0], y1 = yr[1], y2 = yr[2];
        hx  = x0 + x1 + x2;
        hy  = y0 + y1 + y2;
        hxx = x0 * x0 + x1 * x1 + x2 * x2;
        hyy = y0 * y0 + y1 * y1 + y2 * y2;
        hxy = x0 * y0 + x1 * y1 + x2 * y2;
    };

    float ax0, ay0, axx0, ayy0, axy0;
    float ax1, ay1, axx1, ayy1, axy1;
    rowsum(tr0,     ax0, ay0, axx0, ayy0, axy0);
    rowsum(tr0 + 1, ax1, ay1, axx1, ayy1, axy1);

    float acc = 0.0f;
#pragma unroll
    for (int i = 0; i < 16; ++i) {
        float ax2, ay2, axx2, ayy2, axy2;
        rowsum(tr0 + i + 2, ax2, ay2, axx2, ayy2, axy2);

        const float sx  = ax0 + ax1 + ax2;
        const float sy  = ay0 + ay1 + ay2;
        const float sxx = axx0 + axx1 + axx2;
        const float syy = ayy0 + ayy1 + ayy2;
        const float sxy = axy0 + axy1 + axy2;

        const float mx  = sx * inv9;
        const float my  = sy * inv9;
        const float vx  = sxx * inv9 - mx * mx;
        const float vy  = syy * inv9 - my * my;
        const float vxy = sxy * inv9 - mx * my;

        const float num = (2.0f * mx * my + C1f) * (2.0f * vxy + C2f);
        const float den = (mx * mx + my * my + C1f) * (vx + vy + C2f) + EPSf;
        acc += num / den;

        ax0 = ax1; ay0 = ay1; axx0 = axx1; ayy0 = ayy1; axy0 = axy1;
        ax1 = ax2; ay1 = ay2; axx1 = axx2; ayy1 = ayy2; axy1 = axy2;
    }

    // wave32 shuffle reduction -> 8 partials -> one f64 atomic per block.
    float v = acc;
    for (int off = 16; off > 0; off >>= 1) v += __shfl_down(v, off, 32);
    const int lane = tid & 31, wid = tid >> 5;
    if (lane == 0) red[wid] = v;
    __syncthreads();
    if (tid == 0) {
        float s = 0.0f;
#pragma unroll
        for (int w = 0; w < 8; ++w) s += red[w];
        atomicAdd(g_acc, (double)s);   // global_atomic_add_f64
    }
}

__global__ void ssim_init_kernel(double* g_acc) {
    *g_acc = 0.0;
}

__global__ void ssim_final_kernel(const double* __restrict__ g_acc,
                                  float* __restrict__ out) {
    const double mean = *g_acc / (double)((long long)N_IMG * IMG_HW);
    out[0] = 1.0f - (float)mean;
}

extern "C" void kernel_launch(void* const* d_in, const int* in_sizes, int n_in,
                              void* d_out, int out_size, void* d_ws, size_t ws_size,
                              hipStream_t stream) {
    const float* x = (const float*)d_in[0];
    const float* y = (const float*)d_in[1];
    float* out   = (float*)d_out;
    double* acc  = (double*)d_ws;   // 8 bytes of scratch: global f64 accumulator

    hipLaunchKernelGGL(ssim_init_kernel, dim3(1), dim3(1), 0, stream, acc);

    dim3 grid(IMG_W / TILE_W, IMG_H / TILE_H, N_IMG);   // 8 x 32 x 32 = 8192 blocks
    hipLaunchKernelGGL(ssim_main_kernel, grid, dim3(256), 0, stream, x, y, acc);

    hipLaunchKernelGGL(ssim_final_kernel, dim3(1), dim3(1), 0, stream, acc, out);
}